// MODEL_1_52269751992446
// MI455X (gfx1250) — compile-verified
//
#include <hip/hip_runtime.h>

typedef __attribute__((ext_vector_type(2))) float v2f;
typedef __attribute__((ext_vector_type(8))) float v8f;

#define D 512

// C[m,n] = sum_k A[m,k] * W[k,n] + bias[n]
// Grid: one block per 16-row M tile. Block = 256 threads = 8 waves.
// Wave w computes columns [w*64, w*64+64) using 4 x V_WMMA_F32_16X16X4_F32 tiles.
__global__ __launch_bounds__(256) void gemm_wmma_f32(
    const float* __restrict__ A, int lda,
    const float* __restrict__ W,
    const float* __restrict__ bias,
    float* __restrict__ C, int ldc)
{
    const int lane  = threadIdx.x & 31;
    const int wave  = threadIdx.x >> 5;   // 0..7
    const int half  = lane >> 4;          // 0 or 1 (K pair select)
    const int l     = lane & 15;          // row/col within tile
    const int m0    = blockIdx.x * 16;
    const int nbase = wave * 64;

    v8f acc0 = {}, acc1 = {}, acc2 = {}, acc3 = {};

    // A fragment: lanes 0-15 supply (M=l, K=k,k+1); lanes 16-31 supply (M=l, K=k+2,k+3)
    const float* arow = A + (size_t)(m0 + l) * lda + half * 2;
    // B fragment: lanes 0-15 supply (K=k,k+1, N=n); lanes 16-31 (K=k+2,k+3, N=n)
    const float* bcol = W + (size_t)(half * 2) * D + nbase + l;

    for (int k = 0; k < D; k += 4) {
        v2f a;
        a.x = arow[k];
        a.y = arow[k + 1];
        const float* bk = bcol + (size_t)k * D;
        v2f b0, b1, b2, b3;
        b0.x = bk[0];   b0.y = bk[D + 0];
        b1.x = bk[16];  b1.y = bk[D + 16];
        b2.x = bk[32];  b2.y = bk[D + 32];
        b3.x = bk[48];  b3.y = bk[D + 48];
        acc0 = __builtin_amdgcn_wmma_f32_16x16x4_f32(false, a, false, b0, (short)0, acc0, false, false);
        acc1 = __builtin_amdgcn_wmma_f32_16x16x4_f32(false, a, false, b1, (short)0, acc1, false, false);
        acc2 = __builtin_amdgcn_wmma_f32_16x16x4_f32(false, a, false, b2, (short)0, acc2, false, false);
        acc3 = __builtin_amdgcn_wmma_f32_16x16x4_f32(false, a, false, b3, (short)0, acc3, false, false);
    }

    const float bn0 = bias[nbase + l];
    const float bn1 = bias[nbase + 16 + l];
    const float bn2 = bias[nbase + 32 + l];
    const float bn3 = bias[nbase + 48 + l];

    // C/D layout: VGPR r holds (M = r + 8*half, N = l)
    float* crow = C + (size_t)(m0 + half * 8) * ldc + nbase + l;
    #pragma unroll
    for (int r = 0; r < 8; ++r) {
        float* cr = crow + (size_t)r * ldc;
        cr[0]  = acc0[r] + bn0;
        cr[16] = acc1[r] + bn1;
        cr[32] = acc2[r] + bn2;
        cr[48] = acc3[r] + bn3;
    }
}

// dst[rows[e], :] += vals[e] * src[cols[e], :]   (128 lanes per edge, 4 floats each)
__global__ __launch_bounds__(256) void spmm_scatter(
    const float* __restrict__ src,
    float* __restrict__ dst, int ldd,
    const int* __restrict__ rows, const int* __restrict__ cols,
    const float* __restrict__ vals, int E)
{
    const int e = blockIdx.x * 2 + (threadIdx.x >> 7);
    if (e >= E) return;
    const int f = (threadIdx.x & 127) * 4;
    const int c = cols[e];
    const int r = rows[e];
    const float v = vals[e];
    const float4 s = *reinterpret_cast<const float4*>(src + (size_t)c * D + f);
    float* d = dst + (size_t)r * ldd + f;
    unsafeAtomicAdd(d + 0, v * s.x);
    unsafeAtomicAdd(d + 1, v * s.y);
    unsafeAtomicAdd(d + 2, v * s.z);
    unsafeAtomicAdd(d + 3, v * s.w);
}

// Zero a [N x 512] region with row stride ld. One thread per element (N*512 threads).
__global__ __launch_bounds__(256) void zero_rows(float* __restrict__ p, int ld)
{
    const long idx = (long)blockIdx.x * 256 + threadIdx.x;
    const int row = (int)(idx >> 9);
    const int c   = (int)(idx & 511);
    p[(size_t)row * ld + c] = 0.0f;
}

// x1 = a0 * n + s * (1 - n);  a0 has row stride 1024, s and x1 have stride 512
__global__ __launch_bounds__(256) void combine_x1(
    const float* __restrict__ a0,
    const float* __restrict__ s,
    const float* __restrict__ nvec,
    float* __restrict__ x1)
{
    const long idx = (long)blockIdx.x * 256 + threadIdx.x;
    const int row = (int)(idx >> 9);
    const int c   = (int)(idx & 511);
    const float nv = nvec[row];
    x1[idx] = a0[(size_t)row * 1024 + c] * nv + s[idx] * (1.0f - nv);
}

extern "C" void kernel_launch(void* const* d_in, const int* in_sizes, int n_in,
                              void* d_out, int out_size, void* d_ws, size_t ws_size,
                              hipStream_t stream)
{
    const float* A1       = (const float*)d_in[0];   // (N, 513)
    const int*   adj_rows = (const int*)  d_in[1];   // (E,)
    const int*   adj_cols = (const int*)  d_in[2];   // (E,)
    const float* adj_vals = (const float*)d_in[3];   // (E,)
    const float* Lin1     = (const float*)d_in[4];   // (512, 512)
    const float* Lin1b    = (const float*)d_in[5];   // (512,)
    const float* nvec     = (const float*)d_in[6];   // (N, 1)
    const float* W1       = (const float*)d_in[7];
    const float* b1       = (const float*)d_in[8];
    const float* W2       = (const float*)d_in[9];
    const float* b2       = (const float*)d_in[10];
    const float* W3       = (const float*)d_in[11];
    const float* b3       = (const float*)d_in[12];

    float* out = (float*)d_out;                      // (N, 1024) = [x2 | a0]
    const int N = in_sizes[6];                       // 50000
    const int E = in_sizes[1];                       // 1600000

    float* WS0 = (float*)d_ws;                       // (N, 512)
    float* WS1 = WS0 + (size_t)N * D;                // (N, 512)

    const dim3 blk(256);
    const int mtiles = N / 16;                       // 50000 / 16 = 3125 exact
    const int zb = (int)(((long)N * D) / 256);
    const int eb = (E + 1) / 2;

    // a0 = x @ Lin1 + Lin1_bias  -> out[:, 512:1024]
    gemm_wmma_f32<<<mtiles, blk, 0, stream>>>(A1 + 1, D + 1, Lin1, Lin1b, out + D, 2 * D);
    // h1 = x @ W1 + b1 -> WS0
    gemm_wmma_f32<<<mtiles, blk, 0, stream>>>(A1 + 1, D + 1, W1, b1, WS0, D);
    // s1 = spmm(h1) -> WS1
    zero_rows<<<zb, blk, 0, stream>>>(WS1, D);
    spmm_scatter<<<eb, blk, 0, stream>>>(WS0, WS1, D, adj_rows, adj_cols, adj_vals, E);
    // x1 = a0*n + s1*(1-n) -> WS0
    combine_x1<<<zb, blk, 0, stream>>>(out + D, WS1, nvec, WS0);
    // h2 = x1 @ W2 + b2 -> WS1
    gemm_wmma_f32<<<mtiles, blk, 0, stream>>>(WS0, D, W2, b2, WS1, D);
    // t = spmm(h2) -> WS0
    zero_rows<<<zb, blk, 0, stream>>>(WS0, D);
    spmm_scatter<<<eb, blk, 0, stream>>>(WS1, WS0, D, adj_rows, adj_cols, adj_vals, E);
    // h3 = t @ W3 + b3 -> WS1
    gemm_wmma_f32<<<mtiles, blk, 0, stream>>>(WS0, D, W3, b3, WS1, D);
    // x2 = spmm(h3) -> out[:, 0:512]
    zero_rows<<<zb, blk, 0, stream>>>(out, 2 * D);
    spmm_scatter<<<eb, blk, 0, stream>>>(WS1, out, 2 * D, adj_rows, adj_cols, adj_vals, E);
}